// GCNpredict_88957362635436
// MI455X (gfx1250) — compile-verified
//
#include <hip/hip_runtime.h>
#include <hip/hip_bf16.h>

typedef __attribute__((ext_vector_type(2))) float v2f;
typedef __attribute__((ext_vector_type(8))) float v8f;

#define IN_DIM  128
#define OUT_DIM 128
#define LDS_STRIDE 132   // 132 % 64 == 4 -> conflict-free LDS reads across 16 lanes

// ---------------------------------------------------------------------------
// Kernel 1: y = x @ W  (fp32 WMMA, 16x16x4_f32)
// grid.x = ceil(n_nodes/16) blocks, 256 threads (8 waves). Wave w -> N tile w.
// ---------------------------------------------------------------------------
__global__ void gcn_gemm_wmma(const float* __restrict__ x,
                              const float* __restrict__ W,
                              float* __restrict__ y,
                              int n_nodes) {
    __shared__ float sA[16 * LDS_STRIDE];

    const int m0   = blockIdx.x * 16;
    const int tid  = threadIdx.x;
    const int lane = tid & 31;
    const int wave = tid >> 5;
    const bool full_tile = (m0 + 16 <= n_nodes);   // uniform across the block

    // Cooperative load of the 16x128 A strip into LDS (coalesced, 8 floats/thread).
    if (full_tile) {
        for (int i = tid; i < 16 * IN_DIM; i += 256) {
            const int row = i >> 7;
            const int col = i & (IN_DIM - 1);
            sA[row * LDS_STRIDE + col] = x[(m0 + row) * IN_DIM + col];
        }
    } else {
        for (int i = tid; i < 16 * IN_DIM; i += 256) {
            const int row = i >> 7;
            const int col = i & (IN_DIM - 1);
            int grow = m0 + row;
            if (grow >= n_nodes) grow = n_nodes - 1;   // clamp, keep all lanes active
            sA[row * LDS_STRIDE + col] = x[grow * IN_DIM + col];
        }
    }
    __syncthreads();

    // Fragment indexing per CDNA5 32-bit 16x4 A layout:
    //   lanes 0-15  : M = lane,    VGPR0 = K+0, VGPR1 = K+1
    //   lanes 16-31 : M = lane-16, VGPR0 = K+2, VGPR1 = K+3
    // B (4x16) mirrors with N striped across lanes.
    const int half = lane >> 4;        // 0 or 1
    const int mrow = lane & 15;        // A row within tile / B col within tile
    const int coln = wave * 16 + mrow; // global output column

    v8f c = {};
    #pragma unroll 4
    for (int k = 0; k < IN_DIM; k += 4) {
        const int kb = k + half * 2;
        v2f a;
        a.x = sA[mrow * LDS_STRIDE + kb];
        a.y = sA[mrow * LDS_STRIDE + kb + 1];
        v2f b;
        b.x = W[kb * OUT_DIM + coln];
        b.y = W[(kb + 1) * OUT_DIM + coln];
        c = __builtin_amdgcn_wmma_f32_16x16x4_f32(
                /*neg_a=*/false, a, /*neg_b=*/false, b,
                /*c_mod=*/(short)0, c, /*reuse_a=*/false, /*reuse_b=*/false);
    }

    // C/D layout: VGPR r -> M = r (lanes 0-15) or r+8 (lanes 16-31), N = lane&15.
    const int rbase = m0 + half * 8;
    if (full_tile) {
        #pragma unroll
        for (int r = 0; r < 8; ++r)
            y[(rbase + r) * OUT_DIM + coln] = c[r];
    } else {
        #pragma unroll
        for (int r = 0; r < 8; ++r) {
            if (rbase + r < n_nodes)
                y[(rbase + r) * OUT_DIM + coln] = c[r];
        }
    }
}

// ---------------------------------------------------------------------------
// Kernel 2: out[i][c] = bias[c]   (seed the scatter-sum; covers 0-degree nodes)
// ---------------------------------------------------------------------------
__global__ void gcn_bias_init(const float* __restrict__ bias,
                              float* __restrict__ out,
                              int total) {
    const int t = blockIdx.x * blockDim.x + threadIdx.x;
    if (t < total)
        out[t] = bias[t & (OUT_DIM - 1)];
}

// ---------------------------------------------------------------------------
// Kernel 3: out[dst_e][c] += w_e * y[src_e][c]
// 128 threads per edge -> 1 coalesced load + 1 non-returning f32 global atomic
// per thread. Working set (y + out ~ 51 MB) is L2-resident on MI455X (192 MB),
// so both the gather and global_atomic_add_f32 resolve at L2 bandwidth.
// ---------------------------------------------------------------------------
__global__ void gcn_scatter(const float* __restrict__ y,
                            const float* __restrict__ ew,
                            const int* __restrict__ src,
                            const int* __restrict__ dst,
                            float* __restrict__ out,
                            int n_edges) {
    const int t = blockIdx.x * blockDim.x + threadIdx.x;
    const int e = t >> 7;              // edge index
    const int c = t & (OUT_DIM - 1);   // channel
    if (e < n_edges) {
        const float v = ew[e] * y[src[e] * OUT_DIM + c];
        atomicAdd(&out[dst[e] * OUT_DIM + c], v);
    }
}

extern "C" void kernel_launch(void* const* d_in, const int* in_sizes, int n_in,
                              void* d_out, int out_size, void* d_ws, size_t ws_size,
                              hipStream_t stream) {
    const float* x    = (const float*)d_in[0];   // [N, 128]
    const float* ew   = (const float*)d_in[1];   // [E]
    const float* W    = (const float*)d_in[2];   // [128, 128]
    const float* bias = (const float*)d_in[3];   // [128]
    const int*   src  = (const int*)d_in[4];     // [E]
    const int*   dst  = (const int*)d_in[5];     // [E]
    float* out = (float*)d_out;                  // [N, 128]
    float* y   = (float*)d_ws;                   // [N, 128] scratch

    const int n_nodes = in_sizes[0] / IN_DIM;
    const int n_edges = in_sizes[1];

    // 1) y = x @ W (WMMA)
    const int m_tiles = (n_nodes + 15) / 16;
    gcn_gemm_wmma<<<m_tiles, 256, 0, stream>>>(x, W, y, n_nodes);

    // 2) out = bias (broadcast)
    const int total = n_nodes * OUT_DIM;
    gcn_bias_init<<<(total + 255) / 256, 256, 0, stream>>>(bias, out, total);

    // 3) out[dst] += w_e * y[src]
    const long long work = (long long)n_edges * OUT_DIM;
    const int blocks = (int)((work + 255) / 256);
    gcn_scatter<<<blocks, 256, 0, stream>>>(y, ew, src, dst, out, n_edges);
}